// LinearAttention_6322191859739
// MI455X (gfx1250) — compile-verified
//
#include <hip/hip_runtime.h>

// ---------------------------------------------------------------------------
// LinearAttention fused pipeline for gfx1250 (MI455X), fp32 WMMA path.
// Shapes: b=2, c=64, n=32768, heads=4, dim_head=32, hidden=128, qkv=384.
// Memory-bound: 34 MB mandatory HBM traffic; all intermediates L2-resident.
// ---------------------------------------------------------------------------

#define NPOS   32768           // 32*32*32 spatial positions
#define CIN    64
#define HID    128
#define QKVC   384
#define NB     2
#define SCALE_Q 0.17677669529663687f   // 32^-0.5
#define GN_EPS  1e-5f

typedef __attribute__((ext_vector_type(2))) float v2f;
typedef __attribute__((ext_vector_type(8))) float v8f;

__device__ __forceinline__ v8f wmma_f32(v2f a, v2f b, v8f c) {
    // V_WMMA_F32_16X16X4_F32: D(16x16,f32) = A(16x4,f32) x B(4x16,f32) + C
    return __builtin_amdgcn_wmma_f32_16x16x4_f32(
        /*neg_a=*/false, a, /*neg_b=*/false, b,
        /*c_mod=*/(short)0, c, /*reuse_a=*/false, /*reuse_b=*/false);
}

// ---------------------------------------------------------------------------
// Kernel 1: qkv[b][384][n] = w_qkv(384x64) @ x[b](64xn).
// One wave = one 16-row M tile x FOUR 16-col N tiles.  The 16 A fragments
// (16x64 weights striped per the 16x4 f32 layout) are preloaded into
// registers once; the K-loop only streams B (x) rows.
// A layout (16x4 f32): lanes 0-15 row=l K={0,1}; lanes 16-31 row=l K={2,3}.
// B layout (4x16 f32): lanes 0-15 col=l K rows {0,1}; lanes 16-31 rows {2,3}.
// C/D layout: VGPR j -> row j (lanes 0-15) / row j+8 (lanes 16-31).
// ---------------------------------------------------------------------------
__global__ __launch_bounds__(32)
void qkv_gemm(const float* __restrict__ x, const float* __restrict__ wqkv,
              float* __restrict__ qkv) {
    const int lane = threadIdx.x;
    const int half = lane >> 4;        // 0 or 1
    const int l    = lane & 15;
    const int nt   = blockIdx.x * 64;  // 4 N tiles of 16
    const int mt   = blockIdx.y * 16;
    const int b    = blockIdx.z;
    const float* xb = x + (size_t)b * CIN * NPOS;

    v2f af[16];                        // entire 16x64 A tile in registers
    #pragma unroll
    for (int k = 0; k < 16; ++k) {
        const int ka = k * 4 + half * 2;
        af[k].x = wqkv[(mt + l) * CIN + ka];
        af[k].y = wqkv[(mt + l) * CIN + ka + 1];
    }

    v8f acc[4] = {};
    #pragma unroll
    for (int k = 0; k < 16; ++k) {
        const int ka = k * 4 + half * 2;
        const float* r0 = xb + (size_t)ka       * NPOS + nt + l;
        const float* r1 = xb + (size_t)(ka + 1) * NPOS + nt + l;
        #pragma unroll
        for (int t = 0; t < 4; ++t) {
            v2f bb; bb.x = r0[t * 16]; bb.y = r1[t * 16];
            acc[t] = wmma_f32(af[k], bb, acc[t]);
        }
    }

    float* dst = qkv + (size_t)b * QKVC * NPOS;
    #pragma unroll
    for (int t = 0; t < 4; ++t)
        #pragma unroll
        for (int j = 0; j < 8; ++j) {
            const int row = mt + j + half * 8;
            dst[(size_t)row * NPOS + nt + t * 16 + l] = acc[t][j];
        }
}

// ---------------------------------------------------------------------------
// Kernel 2: per (b, k-channel) max and sum(exp) over n  (softmax over axis 3).
// k channels live at qkv channels [128,256).  Streaming scan with prefetch.
// ---------------------------------------------------------------------------
__global__ __launch_bounds__(256)
void k_stats(const float* __restrict__ qkv, float* __restrict__ kmax,
             float* __restrict__ ksum) {
    const int ch = blockIdx.x;              // 0..255 = b*128 + c
    const int b  = ch >> 7, c = ch & 127;
    const float* kp = qkv + (size_t)(b * QKVC + HID + c) * NPOS;
    __shared__ float red[256];

    float m = -1e30f;
    for (int i = threadIdx.x; i < NPOS; i += 256) {
        if (i + 4096 < NPOS) __builtin_prefetch(kp + i + 4096, 0, 1);
        m = fmaxf(m, kp[i]);
    }
    red[threadIdx.x] = m; __syncthreads();
    for (int s = 128; s > 0; s >>= 1) {
        if (threadIdx.x < s) red[threadIdx.x] = fmaxf(red[threadIdx.x], red[threadIdx.x + s]);
        __syncthreads();
    }
    m = red[0]; __syncthreads();

    float sum = 0.f;
    for (int i = threadIdx.x; i < NPOS; i += 256) sum += __expf(kp[i] - m);
    red[threadIdx.x] = sum; __syncthreads();
    for (int s = 128; s > 0; s >>= 1) {
        if (threadIdx.x < s) red[threadIdx.x] += red[threadIdx.x + s];
        __syncthreads();
    }
    if (threadIdx.x == 0) { kmax[ch] = m; ksum[ch] = red[0]; }
}

// ---------------------------------------------------------------------------
// Kernel 3: context[b][h][d][e] = sum_n softmax_k[d][n] * v[e][n].
// One block per (b,h); 8 waves split the n range; each wave accumulates the
// full 32x32 (2x2 tiles of 16x16) with WMMA, A = exp(k - kmax) built in VALU.
// LDS reduce across waves, then divide by ksum[d].
// ---------------------------------------------------------------------------
__global__ __launch_bounds__(256)
void context_gemm(const float* __restrict__ qkv, const float* __restrict__ kmax,
                  const float* __restrict__ ksum, float* __restrict__ ctx) {
    const int bh   = blockIdx.x;            // 0..7
    const int b    = bh >> 2, h = bh & 3;
    const int wave = threadIdx.x >> 5;
    const int lane = threadIdx.x & 31;
    const int half = lane >> 4, l = lane & 15;

    const float* kbase = qkv + (size_t)(b * QKVC + HID      + h * 32) * NPOS;
    const float* vbase = qkv + (size_t)(b * QKVC + 2 * HID  + h * 32) * NPOS;
    const int sbase = b * HID + h * 32;
    const float km0 = kmax[sbase + l];       // rows 0..15 (this lane's row)
    const float km1 = kmax[sbase + 16 + l];  // rows 16..31

    v8f acc[2][2] = {};
    const int chunk = NPOS / 8;
    const int n0 = wave * chunk, n1 = n0 + chunk;
    for (int nn = n0; nn < n1; nn += 4) {
        const int kk = nn + half * 2;
        v2f a0, a1, b0, b1;
        a0.x = __expf(kbase[(size_t)l        * NPOS + kk]     - km0);
        a0.y = __expf(kbase[(size_t)l        * NPOS + kk + 1] - km0);
        a1.x = __expf(kbase[(size_t)(16 + l) * NPOS + kk]     - km1);
        a1.y = __expf(kbase[(size_t)(16 + l) * NPOS + kk + 1] - km1);
        b0.x = vbase[(size_t)l        * NPOS + kk];       // B[k][e=l] = v[l][n]
        b0.y = vbase[(size_t)l        * NPOS + kk + 1];
        b1.x = vbase[(size_t)(16 + l) * NPOS + kk];       // e-tile 1
        b1.y = vbase[(size_t)(16 + l) * NPOS + kk + 1];
        acc[0][0] = wmma_f32(a0, b0, acc[0][0]);
        acc[0][1] = wmma_f32(a0, b1, acc[0][1]);
        acc[1][0] = wmma_f32(a1, b0, acc[1][0]);
        acc[1][1] = wmma_f32(a1, b1, acc[1][1]);
    }

    __shared__ float sctx[8][32][32];       // 32 KB of the 320 KB LDS
    #pragma unroll
    for (int dt = 0; dt < 2; ++dt)
        #pragma unroll
        for (int et = 0; et < 2; ++et)
            #pragma unroll
            for (int j = 0; j < 8; ++j)
                sctx[wave][dt * 16 + j + half * 8][et * 16 + l] = acc[dt][et][j];
    __syncthreads();

    for (int idx = threadIdx.x; idx < 1024; idx += 256) {
        const int d = idx >> 5, e = idx & 31;
        float s = 0.f;
        #pragma unroll
        for (int w = 0; w < 8; ++w) s += sctx[w][d][e];
        ctx[((size_t)(b * 4 + h) * 32 + d) * 32 + e] = s / ksum[sbase + d];
    }
}

// ---------------------------------------------------------------------------
// Kernel 4: per-position q softmax (over the 32 head channels) * SCALE, then
// out[e] = sum_d ctx[h][d][e] * q_sm[d].  Context cached in LDS (broadcast
// reads, conflict-free).  outbuf[b][128][n].
// ---------------------------------------------------------------------------
__global__ __launch_bounds__(256)
void q_apply(const float* __restrict__ qkv, const float* __restrict__ ctx,
             float* __restrict__ outbuf) {
    __shared__ float c[4][32][32];
    const int b = blockIdx.y;
    for (int i = threadIdx.x; i < 4096; i += 256)
        (&c[0][0][0])[i] = ctx[(size_t)b * 4096 + i];
    __syncthreads();

    const int pos = blockIdx.x * 256 + threadIdx.x;
    const float* qb = qkv + (size_t)b * QKVC * NPOS;         // q channels 0..127
    float* ob = outbuf + (size_t)b * HID * NPOS;

    for (int h = 0; h < 4; ++h) {
        float qv[32];
        float m = -1e30f;
        #pragma unroll
        for (int d = 0; d < 32; ++d) {
            qv[d] = qb[(size_t)(h * 32 + d) * NPOS + pos];
            m = fmaxf(m, qv[d]);
        }
        float s = 0.f;
        #pragma unroll
        for (int d = 0; d < 32; ++d) { qv[d] = __expf(qv[d] - m); s += qv[d]; }
        const float inv = SCALE_Q / s;
        for (int e = 0; e < 32; ++e) {
            float acc = 0.f;
            #pragma unroll
            for (int d = 0; d < 32; ++d) acc += c[h][d][e] * qv[d];
            ob[(size_t)(h * 32 + e) * NPOS + pos] = acc * inv;
        }
    }
}

// ---------------------------------------------------------------------------
// Kernel 5: y[b][64][n] = w_out(64x128) @ out + b_out  (WMMA, K=128), plus
// GroupNorm sum / sumsq accumulation (wave shuffle reduce -> one atomic).
// Same 4-N-tile, register-resident-A structure as kernel 1 (32 A fragments).
// y is written into the dead k-region of the qkv buffer.
// ---------------------------------------------------------------------------
__global__ __launch_bounds__(32)
void out_gemm(const float* __restrict__ outbuf, const float* __restrict__ wout,
              const float* __restrict__ bout, float* __restrict__ ybase,
              float* __restrict__ gnsum, float* __restrict__ gnsq) {
    const int lane = threadIdx.x;
    const int half = lane >> 4, l = lane & 15;
    const int nt = blockIdx.x * 64;    // 4 N tiles of 16
    const int mt = blockIdx.y * 16;
    const int b  = blockIdx.z;
    const float* ob = outbuf + (size_t)b * HID * NPOS;

    v2f af[32];                        // entire 16x128 A tile in registers
    #pragma unroll
    for (int k = 0; k < 32; ++k) {
        const int ka = k * 4 + half * 2;
        af[k].x = wout[(mt + l) * HID + ka];
        af[k].y = wout[(mt + l) * HID + ka + 1];
    }

    v8f acc[4] = {};
    #pragma unroll
    for (int k = 0; k < 32; ++k) {
        const int ka = k * 4 + half * 2;
        const float* r0 = ob + (size_t)ka       * NPOS + nt + l;
        const float* r1 = ob + (size_t)(ka + 1) * NPOS + nt + l;
        #pragma unroll
        for (int t = 0; t < 4; ++t) {
            v2f bb; bb.x = r0[t * 16]; bb.y = r1[t * 16];
            acc[t] = wmma_f32(af[k], bb, acc[t]);
        }
    }

    float* yb = ybase + (size_t)b * QKVC * NPOS;   // stride skips to batch b
    float lsum = 0.f, lsq = 0.f;
    #pragma unroll
    for (int t = 0; t < 4; ++t)
        #pragma unroll
        for (int j = 0; j < 8; ++j) {
            const int row = mt + j + half * 8;
            const float v = acc[t][j] + bout[row];
            yb[(size_t)row * NPOS + nt + t * 16 + l] = v;
            lsum += v; lsq += v * v;
        }
    for (int off = 16; off > 0; off >>= 1) {
        lsum += __shfl_down(lsum, off, 32);
        lsq  += __shfl_down(lsq,  off, 32);
    }
    if (lane == 0) {
        atomicAdd(&gnsum[b], lsum);
        atomicAdd(&gnsq[b], lsq);
    }
}

// ---------------------------------------------------------------------------
// Kernel 6: GroupNorm(1 group) finalize: (y-mean)*rsqrt(var+eps)*gamma + beta.
// ---------------------------------------------------------------------------
__global__ __launch_bounds__(256)
void gn_finalize(const float* __restrict__ ybase, const float* __restrict__ gnsum,
                 const float* __restrict__ gnsq, const float* __restrict__ gw,
                 const float* __restrict__ gb, float* __restrict__ out) {
    const size_t idx = (size_t)blockIdx.x * 256 + threadIdx.x;  // < 2*64*NPOS
    const int b   = (int)(idx >> 21);           // 64*NPOS = 2^21
    const int rem = (int)(idx & ((1u << 21) - 1));
    const int o   = rem >> 15;                  // NPOS = 2^15
    const int pos = rem & (NPOS - 1);

    const float count = 64.0f * NPOS;
    const float mean  = gnsum[b] / count;
    const float var   = gnsq[b] / count - mean * mean;
    const float inv   = rsqrtf(var + GN_EPS);
    const float v = ybase[(size_t)b * QKVC * NPOS + (size_t)o * NPOS + pos];
    out[idx] = (v - mean) * inv * gw[o] + gb[o];
}

// ---------------------------------------------------------------------------
extern "C" void kernel_launch(void* const* d_in, const int* in_sizes, int n_in,
                              void* d_out, int out_size, void* d_ws, size_t ws_size,
                              hipStream_t stream) {
    const float* x    = (const float*)d_in[0];
    const float* wqkv = (const float*)d_in[1];
    const float* wout = (const float*)d_in[2];
    const float* bout = (const float*)d_in[3];
    const float* gnw  = (const float*)d_in[4];
    const float* gnb  = (const float*)d_in[5];

    float* ws     = (float*)d_ws;
    float* qkv    = ws;                                        // 25,165,824 f
    float* outbuf = qkv + (size_t)NB * QKVC * NPOS;            //  8,388,608 f
    float* kmax   = outbuf + (size_t)NB * HID * NPOS;          //        256 f
    float* ksum   = kmax + NB * HID;                           //        256 f
    float* ctx    = ksum + NB * HID;                           //      8,192 f
    float* gnsum  = ctx + NB * 4 * 32 * 32;                    //          2 f
    float* gnsq   = gnsum + NB;                                //          2 f
    // y aliases the k channel region [128,192) of qkv (dead after kernel 3)
    float* ybase  = qkv + (size_t)HID * NPOS;

    hipMemsetAsync(gnsum, 0, 2 * NB * sizeof(float), stream);

    qkv_gemm   <<<dim3(NPOS / 64, QKVC / 16, NB), 32,  0, stream>>>(x, wqkv, qkv);
    k_stats    <<<NB * HID, 256, 0, stream>>>(qkv, kmax, ksum);
    context_gemm<<<NB * 4, 256, 0, stream>>>(qkv, kmax, ksum, ctx);
    q_apply    <<<dim3(NPOS / 256, NB), 256, 0, stream>>>(qkv, ctx, outbuf);
    out_gemm   <<<dim3(NPOS / 64, CIN / 16, NB), 32, 0, stream>>>(outbuf, wout, bout,
                                                                  ybase, gnsum, gnsq);
    gn_finalize<<<(NB * CIN * NPOS) / 256, 256, 0, stream>>>(ybase, gnsum, gnsq,
                                                             gnw, gnb, (float*)d_out);
}